// BigramLMParam_41532333752647
// MI455X (gfx1250) — compile-verified
//
#include <hip/hip_runtime.h>

// Z[b,s,:] = W[inputs[b,s], :]  — one-hot matmul is a row gather.
// 4096 rows x 8192 fp32 (32 KB/row). Pure bandwidth problem:
// ~230 MB HBM traffic => ~10us at 23.3 TB/s. WMMA would be ~500x slower.

#define N_COLS    8192
#define THREADS   256
#define ROW_BYTES (N_COLS * 4)          // 32 KB per row
#define CHUNK     (THREADS * 16)        // 4 KB per block-iteration (b128/lane)
#define ITERS     (ROW_BYTES / CHUNK)   // 8

typedef float vfloat4 __attribute__((ext_vector_type(4)));
typedef int   v4i     __attribute__((ext_vector_type(4)));

#if defined(__gfx1250__) &&                                            \
    __has_builtin(__builtin_amdgcn_global_load_async_to_lds_b128) &&   \
    __has_builtin(__builtin_amdgcn_global_store_async_from_lds_b128) && \
    __has_builtin(__builtin_amdgcn_s_wait_asynccnt)
#define USE_ASYNC_LDS 1
#else
#define USE_ASYNC_LDS 0
#endif

__global__ __launch_bounds__(THREADS)
void bigram_gather_rows(const int* __restrict__ tok,
                        const float* __restrict__ W,
                        float* __restrict__ out)
{
    const int bid = blockIdx.x;              // one block per (b,s) position
    const int tid = threadIdx.x;

    // Token id is uniform across the block -> scalar register broadcast.
    const int row = __builtin_amdgcn_readfirstlane(tok[bid]);

    const char* __restrict__ src =
        (const char*)(W + (size_t)row * N_COLS);
    char* __restrict__ dst =
        (char*)(out + (size_t)bid * N_COLS);

#if USE_ASYNC_LDS
    // Stage the row through LDS with the gfx1250 async DMA path:
    // no VGPR round-trip; one ASYNCcnt wait covers all 8 issues.
    __shared__ char lbuf[ROW_BYTES];         // 32 KB

    // Global -> LDS (RT loads: keep W rows hot in L2 for reuse).
#pragma unroll
    for (int i = 0; i < ITERS; ++i) {
        const int off = i * CHUNK + tid * 16;
        __builtin_amdgcn_global_load_async_to_lds_b128(
            (v4i*)(src + off), (v4i*)(lbuf + off),
            /*imm offset*/ 0, /*cpol RT*/ 0);
    }
    __builtin_amdgcn_s_wait_asynccnt(0);

    // LDS -> Global (NT stores: 128 MB streaming output must not
    // evict the reusable W rows from the 192 MB L2).
#pragma unroll
    for (int i = 0; i < ITERS; ++i) {
        const int off = i * CHUNK + tid * 16;
        __builtin_amdgcn_global_store_async_from_lds_b128(
            (v4i*)(dst + off), (v4i*)(lbuf + off),
            /*imm offset*/ 0, /*cpol TH_NT*/ 1);
    }
    __builtin_amdgcn_s_wait_asynccnt(0);
#else
    // Fallback: direct b128 loads + non-temporal b128 stores.
    const vfloat4* __restrict__ s4 = (const vfloat4*)src;
    vfloat4* __restrict__ d4 = (vfloat4*)dst;
#pragma unroll
    for (int i = 0; i < ITERS; ++i) {
        const int e = i * THREADS + tid;
        vfloat4 v = s4[e];
        __builtin_nontemporal_store(v, &d4[e]);
    }
#endif
}

extern "C" void kernel_launch(void* const* d_in, const int* in_sizes, int n_in,
                              void* d_out, int out_size, void* d_ws, size_t ws_size,
                              hipStream_t stream)
{
    (void)n_in; (void)out_size; (void)d_ws; (void)ws_size;
    const int*   tok = (const int*)d_in[0];     // inputs: [8,512] int32
    const float* W   = (const float*)d_in[1];   // W: [8192,8192] fp32
    float*       out = (float*)d_out;           // Z: [8,512,8192] fp32

    const int rows = in_sizes[0];               // 4096 = BATCH*SEQ
    bigram_gather_rows<<<rows, THREADS, 0, stream>>>(tok, W, out);
}